// TemporalEmbedding_12841952215546
// MI455X (gfx1250) — compile-verified
//
#include <hip/hip_runtime.h>

typedef __attribute__((ext_vector_type(2))) float v2f;
typedef __attribute__((ext_vector_type(8))) float v8f;

namespace {
constexpr int kP      = 288;   // K dimension (positions)
constexpr int kD      = 128;   // N dimension (output features)
constexpr int kTileM  = 16;    // rows of B per workgroup
constexpr int kSRow   = 292;   // padded LDS row stride (floats); 292%64=36, gcd=4 -> conflict-free across 16 lanes
constexpr int kThreads = 256;  // 8 wave32
}

__global__ __launch_bounds__(kThreads) void te_fused_wmma_f32(
    const unsigned int* __restrict__ TE,   // int64 data viewed as u32 pairs (we read the low dword)
    const float* __restrict__ wt,          // [163]
    const float* __restrict__ Wlin,        // [D=128, P=288] row-major
    float* __restrict__ out)               // [B, 128]
{
    __shared__ float wt_lds[164];
    __shared__ float s_lds[kTileM * kSRow];

    const int tid = threadIdx.x;
    const int b0  = blockIdx.x * kTileM;

    if (tid < 163) wt_lds[tid] = wt[tid];
    __syncthreads();

    // ---- Phase 1: s[r][p] = wt[m] + wt[12+d] + wt[19+t] for 16x288 tile ----
    // TE flat entry index for (b0+r, p) is (b0*288 + e) with e = r*288+p; each entry is 3 int64.
    const unsigned int* te = TE + (size_t)b0 * kP * 3u * 2u;  // *2: dwords per int64
    #pragma unroll
    for (int i = 0; i < (kTileM * kP) / kThreads; ++i) {
        const int e = i * kThreads + tid;               // 0..4607
        const unsigned vm = te[(e * 3 + 0) * 2];
        const unsigned vd = te[(e * 3 + 1) * 2];
        const unsigned vt = te[(e * 3 + 2) * 2];
        const unsigned m = vm % 12u;
        const unsigned d = vd % 7u;
        const unsigned t = vt % 144u;
        const float s = wt_lds[m] + wt_lds[12u + d] + wt_lds[19u + t];
        const int r = e / kP;
        const int p = e - r * kP;
        s_lds[r * kSRow + p] = s;
    }
    __syncthreads();

    // ---- Phase 2: out[b0:b0+16, n0:n0+16] = S(16x288) x Wlin^T(288x16), f32 WMMA ----
    const int lane = tid & 31;
    const int wave = tid >> 5;        // 0..7 -> N slab
    const int half = lane >> 4;       // 0: K pair {k0,k0+1}; 1: {k0+2,k0+3}
    const int l16  = lane & 15;       // M row (A) / N col (B, C/D)
    const int n0   = wave * 16;

    // B element (k, n) = Wlin[n*288 + k]; per-lane base: column n0+l16, K offset half*2.
    const float* wrow = Wlin + (size_t)(n0 + l16) * kP + half * 2;
    const float* arow = s_lds + l16 * kSRow + half * 2;

    v8f c = {};
    #pragma unroll 4
    for (int k0 = 0; k0 < kP; k0 += 4) {
        v2f a = *(const v2f*)(arow + k0);      // ds_load_b64 from LDS
        v2f b = *(const v2f*)(wrow + k0);      // global_load_b64, L2/WGP$-resident
        // 8 args: (neg_a, A, neg_b, B, c_mod, C, reuse_a, reuse_b)
        c = __builtin_amdgcn_wmma_f32_16x16x4_f32(false, a, false, b, (short)0, c,
                                                  false, false);
    }

    // C/D layout: VGPR i -> lanes 0-15: M=i, lanes 16-31: M=i+8; N = l16.
    float* orow = out + (size_t)b0 * kD + n0 + l16;
    #pragma unroll
    for (int i = 0; i < 8; ++i) {
        orow[(size_t)(i + half * 8) * kD] = c[i];
    }
}

extern "C" void kernel_launch(void* const* d_in, const int* in_sizes, int n_in,
                              void* d_out, int out_size, void* d_ws, size_t ws_size,
                              hipStream_t stream) {
    const unsigned int* TE = (const unsigned int*)d_in[0];  // int64 -> u32 pairs
    const float* wt   = (const float*)d_in[1];              // [163]
    const float* Wlin = (const float*)d_in[2];              // [128, 288]
    float* out = (float*)d_out;                             // [B,1,1,128] flat

    const int Brows = in_sizes[0] / (kP * 3);               // 32768
    dim3 grid(Brows / kTileM);                              // 2048 workgroups
    te_fused_wmma_f32<<<grid, kThreads, 0, stream>>>(TE, wt, Wlin, out);
}